// Mamba_76390288326661
// MI455X (gfx1250) — compile-verified
//
#include <hip/hip_runtime.h>
#include <hip/hip_bf16.h>

// ---------------------------------------------------------------------------
// Mamba block for MI455X (gfx1250), fp32 end-to-end.
//  * Big GEMMs: LDS-blocked (128x64 macro-tile), double-buffered, async
//    global->LDS copies (GLOBAL_LOAD_ASYNC_TO_LDS_B128), fp32 WMMA
//    (V_WMMA_F32_16X16X4_F32) compute, 32x32 register tile per wave.
//  * Small GEMMs: one 16x16 WMMA tile per wave.
//  * Selective scan: register-resident h[16] per channel, B/C staged in LDS
//    64 timesteps at a time.
// ---------------------------------------------------------------------------

typedef float v2f __attribute__((ext_vector_type(2)));
typedef float v8f __attribute__((ext_vector_type(8)));
typedef int   v4i __attribute__((ext_vector_type(4)));

#if __has_builtin(__builtin_amdgcn_global_load_async_to_lds_b128)
#define HAS_ASYNC_LDS 1
#else
#define HAS_ASYNC_LDS 0
#endif

namespace {
constexpr int kDModel = 1024;
constexpr int kDState = 16;
constexpr int kDConv  = 4;
constexpr int kDInner = 2048;
constexpr int kDtRank = 64;
constexpr int kB      = 2;
constexpr int kL      = 2048;
constexpr int kXdbW   = kDtRank + 2 * kDState;  // 96
constexpr int kRows   = kB * kL;                // 4096 (GEMM M)

// Blocked GEMM geometry
constexpr int BM = 128, BN = 64, BK = 16;
constexpr int ASTRIDE = 20;  // 16 k-floats + 4 pad: 16B-aligned rows, 16 banks
}  // namespace

// ---------------------------------------------------------------------------
// Async copy helpers
// ---------------------------------------------------------------------------
__device__ __forceinline__ void copy16_g2l(const float* __restrict__ g,
                                           float* __restrict__ l) {
#if HAS_ASYNC_LDS
  // Prototype (from hipcc diagnostic): (v4i AS1*, v4i AS3*, imm offset, imm cpol)
  __builtin_amdgcn_global_load_async_to_lds_b128(
      (__attribute__((address_space(1))) v4i*)(void*)g,
      (__attribute__((address_space(3))) v4i*)(void*)l, 0, 0);
#else
  *(float4*)l = *(const float4*)g;
#endif
}

__device__ __forceinline__ void wait_async_lds() {
#if HAS_ASYNC_LDS
#if __has_builtin(__builtin_amdgcn_s_wait_asynccnt)
  __builtin_amdgcn_s_wait_asynccnt(0);
#else
  asm volatile("s_wait_asynccnt 0x0" ::: "memory");
#endif
#endif
}

// ---------------------------------------------------------------------------
// fp32 WMMA fragment helpers (ISA 7.12.2 layouts)
// ---------------------------------------------------------------------------
__device__ __forceinline__ v8f wmma4(const v2f& a, const v2f& b, const v8f& c) {
  return __builtin_amdgcn_wmma_f32_16x16x4_f32(false, a, false, b, (short)0, c,
                                               false, false);
}

// C/D 16x16 f32: VGPR r -> row (m0+r) lanes 0-15, row (m0+r+8) lanes 16-31.
__device__ __forceinline__ void store_tile_f32(float* __restrict__ C, int ldc,
                                               int m0, int n0, const v8f& acc) {
  const int lane   = threadIdx.x & 31;
  const int lane16 = lane & 15;
  const int mAdd   = (lane >> 4) << 3;
  float* crow = C + (size_t)(m0 + mAdd) * (size_t)ldc + n0 + lane16;
#pragma unroll
  for (int r = 0; r < 8; ++r) crow[(size_t)r * (size_t)ldc] = acc[r];
}

// ---------------------------------------------------------------------------
// Blocked GEMM: C = A(MxK) @ B(KxN), row-major. Requires M%128==0, N%64==0,
// K%16==0. 256 threads = 8 waves; wave tile 32x32 (four 16x16 accumulators).
// Double-buffered LDS staging, async copies on gfx1250.
// ---------------------------------------------------------------------------
__global__ void __launch_bounds__(256) gemm_f32_wmma_blk(
    const float* __restrict__ A, int lda, const float* __restrict__ B, int ldb,
    float* __restrict__ C, int ldc, int K) {
  __shared__ float As[2][BM * ASTRIDE];  // [m][k], padded
  __shared__ float Bs[2][BK * BN];       // [k][n]

  const int t      = threadIdx.x;
  const int m0blk  = blockIdx.y * BM;
  const int n0blk  = blockIdx.x * BN;
  const int w      = t >> 5;
  const int mw     = (w & 3) << 5;   // wave M offset within block (0..96)
  const int nw     = (w >> 2) << 5;  // wave N offset within block (0 or 32)
  const int lane   = t & 31;
  const int lane16 = lane & 15;
  const int khi    = (lane >> 4) << 1;  // +2 for upper half-wave

  // Cooperative-load mapping (per 128x16 A chunk + 16x64 B chunk)
  const int ar = t >> 1;                 // A row 0..127
  const int ak = (t & 1) << 3;           // A k-offset 0 or 8 (two float4s)
  const int br = t >> 4;                 // B row 0..15
  const int bn = (t & 15) << 2;          // B col offset (float4)
  const float* gA = A + (size_t)(m0blk + ar) * (size_t)lda + ak;
  const float* gB = B + (size_t)br * (size_t)ldb + n0blk + bn;

  auto load_chunk = [&](int k0, int buf) {
    const float* ga = gA + k0;
    copy16_g2l(ga, &As[buf][ar * ASTRIDE + ak]);
    copy16_g2l(ga + 4, &As[buf][ar * ASTRIDE + ak + 4]);
    copy16_g2l(gB + (size_t)k0 * (size_t)ldb, &Bs[buf][br * BN + bn]);
  };

  v8f acc00 = {}, acc01 = {}, acc10 = {}, acc11 = {};

  const int nChunks = K / BK;
  int buf = 0;
  load_chunk(0, 0);
  wait_async_lds();
  __syncthreads();

  for (int c = 0; c < nChunks; ++c) {
    if (c + 1 < nChunks) load_chunk((c + 1) * BK, buf ^ 1);

#pragma unroll
    for (int kk = 0; kk < BK; kk += 4) {
      const int kA = kk + khi;
      const float* Ab = &As[buf][(mw + lane16) * ASTRIDE + kA];
      const float* Bb = &Bs[buf][kA * BN + nw + lane16];
      v2f a0, a1, b0, b1;
      a0.x = Ab[0];            a0.y = Ab[1];
      a1.x = Ab[16 * ASTRIDE]; a1.y = Ab[16 * ASTRIDE + 1];
      b0.x = Bb[0];            b0.y = Bb[BN];
      b1.x = Bb[16];           b1.y = Bb[16 + BN];
      acc00 = wmma4(a0, b0, acc00);
      acc01 = wmma4(a0, b1, acc01);
      acc10 = wmma4(a1, b0, acc10);
      acc11 = wmma4(a1, b1, acc11);
    }

    wait_async_lds();
    __syncthreads();
    buf ^= 1;
  }

  store_tile_f32(C, ldc, m0blk + mw,      n0blk + nw,      acc00);
  store_tile_f32(C, ldc, m0blk + mw,      n0blk + nw + 16, acc01);
  store_tile_f32(C, ldc, m0blk + mw + 16, n0blk + nw,      acc10);
  store_tile_f32(C, ldc, m0blk + mw + 16, n0blk + nw + 16, acc11);
}

// ---------------------------------------------------------------------------
// Simple per-wave GEMM (for small N / small K cases)
// ---------------------------------------------------------------------------
__device__ __forceinline__ v8f wmma_tile_f32(const float* __restrict__ A, int lda,
                                             const float* __restrict__ B, int ldb,
                                             int m0, int n0, int K) {
  const int lane   = threadIdx.x & 31;
  const int lane16 = lane & 15;
  const int khi    = (lane >> 4) << 1;
  const float* arow = A + (size_t)(m0 + lane16) * (size_t)lda;
  const float* bcol = B + n0 + lane16;
  v8f acc = {};
  for (int k = 0; k < K; k += 4) {
    const int kk = k + khi;
    v2f a, b;
    a.x = arow[kk];
    a.y = arow[kk + 1];
    b.x = bcol[(size_t)kk * (size_t)ldb];
    b.y = bcol[(size_t)(kk + 1) * (size_t)ldb];
    acc = wmma4(a, b, acc);
  }
  return acc;
}

__global__ void __launch_bounds__(128) gemm_f32_wmma(
    const float* __restrict__ A, int lda, const float* __restrict__ B, int ldb,
    float* __restrict__ C, int ldc, int M, int N, int K) {
  const int tilesN = N >> 4;
  const int tile   = blockIdx.x * 4 + (threadIdx.x >> 5);
  if (tile >= (M >> 4) * tilesN) return;  // wave-uniform guard
  const int m0 = (tile / tilesN) << 4;
  const int n0 = (tile % tilesN) << 4;
  v8f acc = wmma_tile_f32(A, lda, B, ldb, m0, n0, K);
  store_tile_f32(C, ldc, m0, n0, acc);
}

// GEMM + bias + softplus epilogue: C = softplus(A @ B + bias[col])
__global__ void __launch_bounds__(128) gemm_f32_wmma_softplus(
    const float* __restrict__ A, int lda, const float* __restrict__ B, int ldb,
    const float* __restrict__ bias, float* __restrict__ C, int ldc, int M, int N,
    int K) {
  const int tilesN = N >> 4;
  const int tile   = blockIdx.x * 4 + (threadIdx.x >> 5);
  if (tile >= (M >> 4) * tilesN) return;
  const int m0 = (tile / tilesN) << 4;
  const int n0 = (tile % tilesN) << 4;
  v8f acc = wmma_tile_f32(A, lda, B, ldb, m0, n0, K);
  const int lane   = threadIdx.x & 31;
  const int lane16 = lane & 15;
  const float bn   = bias[n0 + lane16];
  const int mAdd   = (lane >> 4) << 3;
  float* crow = C + (size_t)(m0 + mAdd) * (size_t)ldc + n0 + lane16;
#pragma unroll
  for (int r = 0; r < 8; ++r) {
    float v  = acc[r] + bn;
    float sp = (v > 20.f) ? v : __logf(1.f + __expf(v));  // softplus
    crow[(size_t)r * (size_t)ldc] = sp;
  }
}

// ---------------------------------------------------------------------------
// Causal depthwise conv1d (k=4) + SiLU over xi = x_and_res[..., :D_INNER].
// ---------------------------------------------------------------------------
__global__ void __launch_bounds__(256) conv_silu_kernel(
    const float* __restrict__ xand,     // (4096, 4096), xi in cols [0,2048)
    const float* __restrict__ conv_w,   // (2048, 1, 4)
    const float* __restrict__ conv_b,   // (2048,)
    float* __restrict__ xc) {           // (4096, 2048)
  const int idx = blockIdx.x * blockDim.x + threadIdx.x;
  const int d = idx & (kDInner - 1);
  const int l = (idx >> 11) & (kL - 1);
  const int b = idx >> 22;
  float s = conv_b[d];
#pragma unroll
  for (int j = 0; j < kDConv; ++j) {
    const int ls = l - (kDConv - 1) + j;
    if (ls >= 0) {
      s += conv_w[d * kDConv + j] *
           xand[(size_t)(b * kL + ls) * (size_t)(2 * kDInner) + d];
    }
  }
  const float si = s / (1.f + __expf(-s));  // SiLU
  xc[(size_t)(b * kL + l) * (size_t)kDInner + d] = si;
}

// ---------------------------------------------------------------------------
// Selective scan. One thread per channel (b, d); h[16] in registers.
// B_t / C_t staged in LDS 64 timesteps at a time (one barrier pair / 64 steps).
// 64 threads per block, 64 blocks -> recurrence spread over many WGPs.
// ---------------------------------------------------------------------------
__global__ void __launch_bounds__(64) selective_scan_kernel(
    const float* __restrict__ xdb,     // (rows, 96): [dt(64) | B(16) | C(16)]
    const float* __restrict__ delta,   // (rows, 2048)
    const float* __restrict__ xc,      // (rows, 2048)  u
    const float* __restrict__ xand,    // (rows, 4096)  res in cols [2048,4096)
    const float* __restrict__ A_log,   // (2048, 16)
    const float* __restrict__ D_param, // (2048,)
    float* __restrict__ yg) {          // (rows, 2048)
  constexpr int CL = 64;  // timesteps staged per LDS refill
  const int b = blockIdx.x >> 5;                        // 64 blocks: 2 b x 32
  const int d = ((blockIdx.x & 31) << 6) + threadIdx.x; // 64 channels per block

  __shared__ float sBC[CL * 2 * kDState];  // [step][B(16)|C(16)] = 8 KB

  float a_n[kDState], h[kDState];
#pragma unroll
  for (int n = 0; n < kDState; ++n) {
    a_n[n] = -__expf(A_log[(size_t)d * kDState + n]);  // A = -exp(A_log)
    h[n]   = 0.f;
  }
  const float Dd = D_param[d];

  for (int lc = 0; lc < kL; lc += CL) {
    __syncthreads();  // previous chunk fully consumed
    // 64 threads x 32 elements = 2048 floats
#pragma unroll 4
    for (int j = 0; j < 32; ++j) {
      const int idx  = threadIdx.x * 32 + j;
      const int step = idx >> 5;
      const int col  = idx & 31;
      sBC[idx] = xdb[((size_t)b * kL + lc + step) * kXdbW + kDtRank + col];
    }
    __syncthreads();

    for (int l = lc; l < lc + CL; ++l) {
      const size_t row = (size_t)b * kL + l;
      const float* bc  = &sBC[(l - lc) * 2 * kDState];
      const float dl = delta[row * kDInner + d];
      const float u  = xc[row * kDInner + d];
      const float du = dl * u;
      float y = 0.f;
#pragma unroll
      for (int n = 0; n < kDState; ++n) {
        const float abar = __expf(dl * a_n[n]);
        h[n] = abar * h[n] + du * bc[n];
        y += h[n] * bc[kDState + n];
      }
      y += Dd * u;
      const float r = xand[row * (size_t)(2 * kDInner) + kDInner + d];
      y *= r / (1.f + __expf(-r));  // y * silu(res)
      yg[row * kDInner + d] = y;
    }
  }
}

// ---------------------------------------------------------------------------
// Launcher
// ---------------------------------------------------------------------------
extern "C" void kernel_launch(void* const* d_in, const int* in_sizes, int n_in,
                              void* d_out, int out_size, void* d_ws, size_t ws_size,
                              hipStream_t stream) {
  const float* x      = (const float*)d_in[0];  // (2, 2048, 1024)
  const float* W_in   = (const float*)d_in[1];  // (1024, 4096)
  const float* conv_w = (const float*)d_in[2];  // (2048, 1, 4)
  const float* conv_b = (const float*)d_in[3];  // (2048,)
  const float* W_x    = (const float*)d_in[4];  // (2048, 96)
  const float* W_dt   = (const float*)d_in[5];  // (64, 2048)
  const float* b_dt   = (const float*)d_in[6];  // (2048,)
  const float* A_log  = (const float*)d_in[7];  // (2048, 16)
  const float* D_par  = (const float*)d_in[8];  // (2048,)
  const float* W_out  = (const float*)d_in[9];  // (2048, 1024)
  float* out = (float*)d_out;                   // (2, 2048, 1024)

  // Workspace layout (bytes)
  char* ws = (char*)d_ws;
  float* xand  = (float*)(ws);                          // 4096*4096 f32 = 64 MB
  float* xc    = (float*)(ws + (size_t)67108864);       // 4096*2048 f32 = 32 MB
  float* xdb   = (float*)(ws + (size_t)100663296);      // 4096*96   f32 = 1.5 MB
  float* delta = (float*)(ws + (size_t)102236160);      // 4096*2048 f32 = 32 MB
  float* yg    = (float*)(ws + (size_t)135790592);      // 4096*2048 f32 = 32 MB

  // 1) x_and_res = x @ W_in            M=4096 N=4096 K=1024 (blocked)
  {
    dim3 grid((2 * kDInner) / BN, kRows / BM);  // (64, 32)
    gemm_f32_wmma_blk<<<grid, 256, 0, stream>>>(x, kDModel, W_in, 2 * kDInner,
                                                xand, 2 * kDInner, kDModel);
  }

  // 2) causal depthwise conv + SiLU -> xc
  conv_silu_kernel<<<(kB * kL * kDInner) / 256, 256, 0, stream>>>(xand, conv_w,
                                                                  conv_b, xc);

  // 3) x_db = xc @ W_x                 M=4096 N=96 K=2048 (simple)
  {
    const int tiles = (kRows / 16) * (kXdbW / 16);  // 1536
    gemm_f32_wmma<<<tiles / 4, 128, 0, stream>>>(xc, kDInner, W_x, kXdbW, xdb,
                                                 kXdbW, kRows, kXdbW, kDInner);
  }

  // 4) delta = softplus(dt @ W_dt + b_dt)   M=4096 N=2048 K=64 (simple+epilogue)
  {
    const int tiles = (kRows / 16) * (kDInner / 16);  // 32768
    gemm_f32_wmma_softplus<<<tiles / 4, 128, 0, stream>>>(
        xdb, kXdbW, W_dt, kDInner, b_dt, delta, kDInner, kRows, kDInner, kDtRank);
  }

  // 5) selective scan + D skip + SiLU(res) gate -> yg
  selective_scan_kernel<<<64, 64, 0, stream>>>(xdb, delta, xc, xand, A_log,
                                               D_par, yg);

  // 6) out = yg @ W_out                M=4096 N=1024 K=2048 (blocked)
  {
    dim3 grid(kDModel / BN, kRows / BM);  // (16, 32)
    gemm_f32_wmma_blk<<<grid, 256, 0, stream>>>(yg, kDInner, W_out, kDModel, out,
                                                kDModel, kDInner);
  }
}